// LbpPooling2D_2714419331940
// MI455X (gfx1250) — compile-verified
//
#include <hip/hip_runtime.h>

// ---------------- problem geometry (from reference) ----------------
#define IH   512
#define IW   512
#define NCH  3
#define WC   1536           // IW*NCH floats per input row (contiguous)
#define OH   256
#define OW   256
#define OWC  768            // OW*NCH floats per output row

// ---------------- tiling ----------------
#define TI        4         // output rows per block
#define TJ        32        // output cols per block
#define TQ        96        // TJ*NCH combined outputs per row
#define NTHREADS  384       // 12 wave32 waves; 384 = TI*TQ exactly
#define IN_ROWS   9         // 2*TI + 1
#define IN_COLS   195       // (2*TJ + 1) * NCH
#define LDS_STRIDE 196      // padded row stride (floats)
#define BULK4     48        // 48 float4 = cols 0..191 (always in col range)

// ---------------- CDNA5 async global->LDS path (probe-guarded) ----------------
#if defined(__HIP_DEVICE_COMPILE__)
#if __has_builtin(__builtin_amdgcn_global_load_async_to_lds_b128) && \
    __has_builtin(__builtin_amdgcn_global_load_async_to_lds_b32)
#define USE_ASYNC_LDS 1
#endif
#endif

// Exact pointer types per clang's builtin prototypes ("V4i*1", "V4i*3", "i*1", "i*3")
typedef int v4i_t __attribute__((vector_size(16)));
typedef __attribute__((address_space(1))) v4i_t* g128p;
typedef __attribute__((address_space(3))) v4i_t* l128p;
typedef __attribute__((address_space(1))) int*   g32p;
typedef __attribute__((address_space(3))) int*   l32p;

__device__ __forceinline__ void async_wait_zero() {
#if defined(__HIP_DEVICE_COMPILE__)
#if __has_builtin(__builtin_amdgcn_s_wait_asynccnt)
  __builtin_amdgcn_s_wait_asynccnt(0);
#else
  asm volatile("s_wait_asynccnt 0x0" ::: "memory");
#endif
#endif
}

__global__ __launch_bounds__(NTHREADS)
void lbp_pool_kernel(const float* __restrict__ x, float* __restrict__ out) {
  __shared__ float tile[IN_ROWS * LDS_STRIDE];

  const int tid = threadIdx.x;
  const int jt  = blockIdx.x;          // 0 .. OW/TJ-1
  const int it  = blockIdx.y;          // 0 .. OH/TI-1
  const int b   = blockIdx.z;          // batch
  const int i0  = it * TI;             // first output row
  const int j0  = jt * TJ;             // first output col
  const int g0  = 6 * j0;              // first input combined col (floats)

  // ---------------- stage the input tile into LDS ----------------
#if USE_ASYNC_LDS
  // bulk: IN_ROWS rows x 48 float4 (cols 0..191) -- always within row width,
  // 16B-aligned on both global and LDS sides.
  for (int t = tid; t < IN_ROWS * BULK4; t += NTHREADS) {
    const int r    = t / BULK4;
    const int c4   = t % BULK4;
    const int grow = 2 * i0 + r;
    float* lp = &tile[r * LDS_STRIDE + c4 * 4];
    if (grow < IH) {
      float* gp = (float*)(x + (long)(b * IH + grow) * WC + g0 + c4 * 4);
      __builtin_amdgcn_global_load_async_to_lds_b128((g128p)gp, (l128p)lp, 0, 0);
    } else {
      lp[0] = 0.0f; lp[1] = 0.0f; lp[2] = 0.0f; lp[3] = 0.0f;   // bottom zero-pad
    }
  }
  // tail: cols 192..194 (3 floats per row); right edge may zero-pad
  for (int t = tid; t < IN_ROWS * 3; t += NTHREADS) {
    const int r    = t / 3;
    const int cc   = 192 + (t % 3);
    const int grow = 2 * i0 + r;
    const int gcol = g0 + cc;
    float* lp = &tile[r * LDS_STRIDE + cc];
    if (grow < IH && gcol < WC) {
      float* gp = (float*)(x + (long)(b * IH + grow) * WC + gcol);
      __builtin_amdgcn_global_load_async_to_lds_b32((g32p)gp, (l32p)lp, 0, 0);
    } else {
      *lp = 0.0f;
    }
  }
  async_wait_zero();
#else
  for (int t = tid; t < IN_ROWS * BULK4; t += NTHREADS) {
    const int r    = t / BULK4;
    const int c4   = t % BULK4;
    const int grow = 2 * i0 + r;
    float4 v = make_float4(0.f, 0.f, 0.f, 0.f);
    if (grow < IH)
      v = *(const float4*)(x + (long)(b * IH + grow) * WC + g0 + c4 * 4);
    *(float4*)&tile[r * LDS_STRIDE + c4 * 4] = v;
  }
  for (int t = tid; t < IN_ROWS * 3; t += NTHREADS) {
    const int r    = t / 3;
    const int cc   = 192 + (t % 3);
    const int grow = 2 * i0 + r;
    const int gcol = g0 + cc;
    float v = 0.0f;
    if (grow < IH && gcol < WC) v = x[(long)(b * IH + grow) * WC + gcol];
    tile[r * LDS_STRIDE + cc] = v;
  }
#endif
  __syncthreads();

  // ---------------- compute: one output element per thread ----------------
  const int qx = tid % TQ;             // combined col in tile (0..95)
  const int iy = tid / TQ;             // row in tile (0..3)
  const int jl = qx / 3;               // local output col
  const int ch = qx % 3;               // channel

  const int base = (2 * iy) * LDS_STRIDE + 6 * jl + ch;
  const float p0 = tile[base + 0];
  const float p1 = tile[base + 3];
  const float p2 = tile[base + 6];
  const float p3 = tile[base + LDS_STRIDE + 0];
  const float p4 = tile[base + LDS_STRIDE + 3];   // center
  const float p5 = tile[base + LDS_STRIDE + 6];
  const float p6 = tile[base + 2 * LDS_STRIDE + 0];
  const float p7 = tile[base + 2 * LDS_STRIDE + 3];
  const float p8 = tile[base + 2 * LDS_STRIDE + 6];

  // MASK = {64,128,1,32,0,2,16,8,4} for k = 3*di+dj
  float s = 0.0f;
  s += (p0 >= p4) ? 64.0f  : 0.0f;
  s += (p1 >= p4) ? 128.0f : 0.0f;
  s += (p2 >= p4) ? 1.0f   : 0.0f;
  s += (p3 >= p4) ? 32.0f  : 0.0f;
  // center contributes MASK[4] = 0
  s += (p5 >= p4) ? 2.0f   : 0.0f;
  s += (p6 >= p4) ? 16.0f  : 0.0f;
  s += (p7 >= p4) ? 8.0f   : 0.0f;
  s += (p8 >= p4) ? 4.0f   : 0.0f;

  const float sum  = ((p0 + p1) + (p2 + p3)) + ((p4 + p5) + (p6 + p7)) + p8;
  const float mean = sum * (1.0f / 9.0f);
  float d, var = 0.0f;
  d = p0 - mean; var += d * d;
  d = p1 - mean; var += d * d;
  d = p2 - mean; var += d * d;
  d = p3 - mean; var += d * d;
  d = p4 - mean; var += d * d;
  d = p5 - mean; var += d * d;
  d = p6 - mean; var += d * d;
  d = p7 - mean; var += d * d;
  d = p8 - mean; var += d * d;
  const float stdv = sqrtf(var * (1.0f / 9.0f));

  const float mx = fmaxf(fmaxf(fmaxf(p0, p1), fmaxf(p2, p3)),
                         fmaxf(fmaxf(p4, p5), fmaxf(p6, fmaxf(p7, p8))));

  const float res = (s * stdv + (1.0f - s) * mx) * (1.0f / 255.0f);

  // coalesced, streaming (non-temporal) store
  float* dst = out + (long)(b * OH + (i0 + iy)) * OWC + 3 * j0 + qx;
#if defined(__HIP_DEVICE_COMPILE__) && __has_builtin(__builtin_nontemporal_store)
  __builtin_nontemporal_store(res, dst);
#else
  *dst = res;
#endif
}

extern "C" void kernel_launch(void* const* d_in, const int* in_sizes, int n_in,
                              void* d_out, int out_size, void* d_ws, size_t ws_size,
                              hipStream_t stream) {
  const float* x = (const float*)d_in[0];
  float* out = (float*)d_out;
  const int B = in_sizes[0] / (IH * IW * NCH);   // 64 for the reference shapes
  dim3 grid(OW / TJ, OH / TI, B);
  dim3 block(NTHREADS);
  hipLaunchKernelGGL(lbp_pool_kernel, grid, block, 0, stream, x, out);
}